// ObjectLoss_45432164057703
// MI455X (gfx1250) — compile-verified
//
#include <hip/hip_runtime.h>

// ---------------------------------------------------------------------------
// ObjectLoss on MI455X (gfx1250, wave32)
//
// Used inputs: pred[N,6] (d_in[3]), particle_id[N] (d_in[5]),
//              track_params[N,6] (d_in[6]), reconstructable[N] (d_in[7]).
//
// Roofline: 5M hits * (24+24+4+4) B = ~280 MB read-once -> ~12 us @ 23.3 TB/s.
// Compute ~35 MFLOP (negligible) -> pure streaming + scatter problem.
// Bucket arrays (50k f32 sums + 50k u32 counts = 400 KB) stay L2-resident;
// the streamed rows use non-temporal loads so they do not displace them.
// One hit per thread (5M threads): latency hiding via occupancy.
//
// objloss_main is FIRST in this TU so the disasm snippet shows its body:
// verify global_atomic_add_f32 (hardware FP atomic, not a CAS loop),
// global_atomic_add_u32, and nt-flagged global_load_b64s.
// ---------------------------------------------------------------------------

#define SEG_P 50000

typedef float vf2 __attribute__((ext_vector_type(2)));

// --- hot kernel: streaming pass, scatter atomics into L2-resident buckets --
__global__ __launch_bounds__(256) static void objloss_main(
    const float* __restrict__ pred,
    const float* __restrict__ tp,
    const int* __restrict__ pid,
    const int* __restrict__ rec,
    float* __restrict__ sums,
    unsigned* __restrict__ counts,
    int n) {
  int i = blockIdx.x * blockDim.x + threadIdx.x;
  if (i >= n) return;

  int p = __builtin_nontemporal_load(pid + i);
  int r = __builtin_nontemporal_load(rec + i);
  if ((p > 0) & (r > 0)) {
    // Rows are 24 B -> three always-8-byte-aligned float2 loads each.
    const vf2* pp = (const vf2*)(pred + (size_t)i * 6);
    const vf2* tt = (const vf2*)(tp + (size_t)i * 6);
    vf2 a0 = __builtin_nontemporal_load(pp + 0);
    vf2 a1 = __builtin_nontemporal_load(pp + 1);
    vf2 a2 = __builtin_nontemporal_load(pp + 2);
    vf2 b0 = __builtin_nontemporal_load(tt + 0);
    vf2 b1 = __builtin_nontemporal_load(tt + 1);
    vf2 b2 = __builtin_nontemporal_load(tt + 2);
    vf2 d0 = a0 - b0;
    vf2 d1 = a1 - b1;
    vf2 d2 = a2 - b2;
    float mse = d0.x * d0.x + d0.y * d0.y +
                d1.x * d1.x + d1.y * d1.y +
                d2.x * d2.x + d2.y * d2.y;
    // Hardware FP atomic + u32 atomic into the L2-resident bucket region.
    unsafeAtomicAdd(&sums[p], mse);
    atomicAdd(&counts[p], 1u);
  }
}

// --- kernel: zero sums[P] (f32) and counts[P] (u32) ------------------------
__global__ __launch_bounds__(256) static void objloss_zero(
    unsigned* __restrict__ ws, int n) {
  int i = blockIdx.x * blockDim.x + threadIdx.x;
  if (i < n) ws[i] = 0u;
}

// --- kernel: reduce 50k buckets -> scalar ----------------------------------
__global__ __launch_bounds__(1024) static void objloss_reduce(
    const float* __restrict__ sums,
    const unsigned* __restrict__ counts,
    float* __restrict__ out) {
  __shared__ float s_m[1024];
  __shared__ float s_k[1024];
  float m = 0.0f;
  float k = 0.0f;
  for (int p = threadIdx.x; p < SEG_P; p += 1024) {
    unsigned c = counts[p];
    if (c > 0u) {
      m += sums[p] / (float)c;
      k += 1.0f;
    }
  }
  s_m[threadIdx.x] = m;
  s_k[threadIdx.x] = k;
  __syncthreads();
  for (int off = 512; off > 0; off >>= 1) {
    if (threadIdx.x < (unsigned)off) {
      s_m[threadIdx.x] += s_m[threadIdx.x + off];
      s_k[threadIdx.x] += s_k[threadIdx.x + off];
    }
    __syncthreads();
  }
  if (threadIdx.x == 0) {
    float K = s_k[0];
    out[0] = (K > 0.0f) ? (100.0f * s_m[0] / K) : 0.0f;
  }
}

// ---------------------------------------------------------------------------
extern "C" void kernel_launch(void* const* d_in, const int* in_sizes, int n_in,
                              void* d_out, int out_size, void* d_ws,
                              size_t ws_size, hipStream_t stream) {
  // setup_inputs() order: W, beta, H, pred, Y, particle_id, track_params,
  // reconstructable.  Only indices 3, 5, 6, 7 are used by the reference.
  const float* pred = (const float*)d_in[3];
  const int* pid = (const int*)d_in[5];
  const float* tp = (const float*)d_in[6];
  const int* rec = (const int*)d_in[7];
  const int n = in_sizes[5];  // N (hits)

  float* sums = (float*)d_ws;
  unsigned* counts = (unsigned*)(sums + SEG_P);

  // Zero the bucket arrays every call (graph replays must be idempotent).
  const int zn = 2 * SEG_P;
  objloss_zero<<<(zn + 255) / 256, 256, 0, stream>>>((unsigned*)d_ws, zn);

  // One hit per thread: 5M threads saturate HBM via occupancy.
  objloss_main<<<(n + 255) / 256, 256, 0, stream>>>(pred, tp, pid, rec, sums,
                                                    counts, n);

  // Single-block final reduce over 50k buckets (reads ~400 KB from L2).
  objloss_reduce<<<1, 1024, 0, stream>>>(sums, counts, (float*)d_out);
}